// OptimalSVDDenoiser_80427557585212
// MI455X (gfx1250) — compile-verified
//
#include <hip/hip_runtime.h>

// ---------------------------------------------------------------------------
// OptimalSVDDenoiser for MI455X (gfx1250, wave32, WMMA f32 16x16x4)
//
// Per batch (2048 total): Xc = center(X[b]) (256x64), G = Xc^T Xc (WMMA),
// Jacobi eigensolve G = V L V^T (VALU), MP-optimal shrink of s = sqrt(L),
// M = V diag(s'/s) V^T (WMMA), Y = Xc*M + m, scaled by 1/(2+rank) (WMMA).
// GEMM stages reuse the A fragment across multiple B tiles (independent
// accumulators -> no WMMA D->A/B RAW hazard stalls, ~40% fewer ds loads).
// ---------------------------------------------------------------------------

typedef float v2f __attribute__((ext_vector_type(2)));
typedef float v8f __attribute__((ext_vector_type(8)));

#define B_BATCH 2048
#define NROW    256     // N = 16*16
#define TCOL    64      // T
#define LDX     68      // padded LDS stride (floats); 68 % 64 == 4
#define LDG     68
#define NSWEEP  8
// median of Marchenko-Pastur distribution, beta = T/N = 0.25 (precomputed
// with the same numeric integral the reference uses at trace time)
#define MP_MEDIAN_B025 0.92442f

static __device__ __forceinline__ v8f wmma4(v2f a, v2f b, v8f c) {
  // 8 args: (neg_a, A, neg_b, B, c_mod, C, reuse_a, reuse_b)
  return __builtin_amdgcn_wmma_f32_16x16x4_f32(false, a, false, b, (short)0, c,
                                               false, false);
}

__global__ __launch_bounds__(256) void
svd_denoise_kernel(const float* __restrict__ x, float* __restrict__ out_x,
                   float* __restrict__ out_w) {
  extern __shared__ float smem[];
  float* Xc   = smem;                  // 256*68 centered data
  float* G    = Xc + NROW * LDX;       // 64*68 Gram matrix, later holds M
  float* V    = G + TCOL * LDG;        // 64*68 eigenvectors
  float* mrow = V + TCOL * LDG;        // 256 row means

  __shared__ float csc[32], css[32];
  __shared__ int   pps[32], qqs[32];
  __shared__ float sv[64], gg[64];
  __shared__ float med_sh, scale_sh, wgt_sh;
  __shared__ int   cnt_sh;

  const int tid  = threadIdx.x;
  const int lane = tid & 31;
  const int wave = tid >> 5;
  const int kh   = lane >> 4;   // which K-pair half of the wave
  const int l16  = lane & 15;
  const int b    = blockIdx.x;

  // ---- 1) load row, compute mean over T, store centered row to LDS --------
  {
    const float4* src =
        reinterpret_cast<const float4*>(x + (size_t)b * NROW * TCOL + tid * TCOL);
    float4 v[16];
    float s = 0.f;
#pragma unroll
    for (int i = 0; i < 16; ++i) {
      v[i] = src[i];
      s += v[i].x + v[i].y + v[i].z + v[i].w;
    }
    const float mean = s * (1.0f / TCOL);
    mrow[tid] = mean;
#pragma unroll
    for (int i = 0; i < 16; ++i) {
      Xc[tid * LDX + 4 * i + 0] = v[i].x - mean;
      Xc[tid * LDX + 4 * i + 1] = v[i].y - mean;
      Xc[tid * LDX + 4 * i + 2] = v[i].z - mean;
      Xc[tid * LDX + 4 * i + 3] = v[i].w - mean;
    }
  }
  __syncthreads();

  // ---- 2) G = Xc^T Xc : wave owns row-tile ti = wave>>1, two col tiles ----
  {
    const int ti  = wave >> 1;
    const int tj0 = (wave & 1) * 2;           // col tiles tj0, tj0+1
    v8f acc0 = {0.f, 0.f, 0.f, 0.f, 0.f, 0.f, 0.f, 0.f};
    v8f acc1 = acc0;
    for (int k0 = 0; k0 < NROW; k0 += 4) {
      const int r = k0 + 2 * kh;
      v2f a, b0, b1;
      a[0]  = Xc[(r    ) * LDX + ti * 16 + l16];          // A[i][k] = Xc[k][i]
      a[1]  = Xc[(r + 1) * LDX + ti * 16 + l16];
      b0[0] = Xc[(r    ) * LDX + tj0 * 16 + l16];         // B[k][j] = Xc[k][j]
      b0[1] = Xc[(r + 1) * LDX + tj0 * 16 + l16];
      b1[0] = Xc[(r    ) * LDX + (tj0 + 1) * 16 + l16];
      b1[1] = Xc[(r + 1) * LDX + (tj0 + 1) * 16 + l16];
      acc0 = wmma4(a, b0, acc0);
      acc1 = wmma4(a, b1, acc1);
    }
#pragma unroll
    for (int v = 0; v < 8; ++v) {
      const int row = ti * 16 + v + 8 * kh;
      G[row * LDG + tj0 * 16 + l16]       = acc0[v];
      G[row * LDG + (tj0 + 1) * 16 + l16] = acc1[v];
    }
  }

  // ---- 3) V = I ------------------------------------------------------------
  for (int e = tid; e < 64 * 64; e += 256) {
    const int i = e >> 6, j = e & 63;
    V[i * LDG + j] = (i == j) ? 1.f : 0.f;
  }
  __syncthreads();

  // ---- 4) parallel cyclic Jacobi (round-robin tournament pairing) ---------
  for (int sweep = 0; sweep < NSWEEP; ++sweep) {
    for (int r = 0; r < 63; ++r) {
      if (tid < 32) {
        int p, q;
        if (tid == 0) { p = 63; q = r % 63; }
        else          { p = (r + tid) % 63; q = (r + 63 - tid) % 63; }
        const float app = G[p * LDG + p];
        const float aqq = G[q * LDG + q];
        const float apq = G[p * LDG + q];
        float c = 1.f, s = 0.f;
        if (fabsf(apq) > 1e-12f) {
          const float tau = (aqq - app) / (2.f * apq);
          const float tt  = copysignf(1.f, tau) /
                            (fabsf(tau) + sqrtf(1.f + tau * tau));
          c = rsqrtf(1.f + tt * tt);
          s = tt * c;
        }
        pps[tid] = p; qqs[tid] = q; csc[tid] = c; css[tid] = s;
      }
      __syncthreads();
      {  // row phase: G <- J^T G  (32 pairs x 64 cols, 8 cols per thread)
        const int pr = tid >> 3;
        const int jb = (tid & 7) * 8;
        const int p = pps[pr], q = qqs[pr];
        const float c = csc[pr], s = css[pr];
#pragma unroll
        for (int j = 0; j < 8; ++j) {
          const float gp = G[p * LDG + jb + j], gq = G[q * LDG + jb + j];
          G[p * LDG + jb + j] = c * gp - s * gq;
          G[q * LDG + jb + j] = s * gp + c * gq;
        }
      }
      __syncthreads();
      {  // col phase: G <- G J, V <- V J.  row = (tid&7) + 8*i makes the 8
         // lanes of a pair hit consecutive rows -> distinct LDS banks
         // (stride 68 == 4 mod 64 -> banks p+4*row, 4 apart, no conflict).
        const int pr = tid >> 3;
        const int r0 = tid & 7;
        const int p = pps[pr], q = qqs[pr];
        const float c = csc[pr], s = css[pr];
#pragma unroll
        for (int i = 0; i < 8; ++i) {
          const int row = r0 + 8 * i;
          const float gp = G[row * LDG + p], gq = G[row * LDG + q];
          G[row * LDG + p] = c * gp - s * gq;
          G[row * LDG + q] = s * gp + c * gq;
          const float vp = V[row * LDG + p], vq = V[row * LDG + q];
          V[row * LDG + p] = c * vp - s * vq;
          V[row * LDG + q] = s * vp + c * vq;
        }
      }
      __syncthreads();
    }
  }

  // ---- 5) singular values, lower-median, MP-optimal shrink ----------------
  if (tid < 64) sv[tid] = sqrtf(fmaxf(G[tid * LDG + tid], 0.f));
  if (tid == 0) cnt_sh = 0;
  __syncthreads();
  if (tid < 64) {  // unique rank via index tie-break; rank 31 = lower median
    int rank = 0;
    for (int j = 0; j < 64; ++j) {
      const float o = sv[j];
      rank += (o < sv[tid]) || (o == sv[tid] && j < tid);
    }
    if (rank == 31) med_sh = sv[tid];
  }
  __syncthreads();
  if (tid == 0) scale_sh = med_sh * (1.0f / sqrtf(MP_MEDIAN_B025));
  __syncthreads();
  if (tid < 64) {
    const float beta  = 0.25f;
    const float scale = scale_sh;
    const float sval  = sv[tid];
    const float y     = (scale != 0.f) ? sval / scale : 0.f;
    const float t0    = y * y - beta - 1.0f;
    const float num   = sqrtf(fmaxf(t0 * t0 - 4.0f * beta, 0.0f));
    float sh = (y > 0.f) ? (num / y) * scale : 0.f;
    if (!(fabsf(sh) < 3.4e38f)) sh = 0.f;  // nan_to_num (NaN/Inf -> 0)
    gg[tid] = (sval > 0.f) ? sh / sval : 0.f;
    if (sh > 0.f) atomicAdd(&cnt_sh, 1);
  }
  __syncthreads();
  if (tid == 0) wgt_sh = 1.0f / (3.0f + (float)cnt_sh);  // 1/(2 + (cnt+1))
  __syncthreads();

  // ---- 6) M = (V diag(gg)) V^T, written over G (WMMA, A-frag reuse) -------
  {
    const int ti  = wave >> 1;
    const int tj0 = (wave & 1) * 2;
    v8f acc0 = {0.f, 0.f, 0.f, 0.f, 0.f, 0.f, 0.f, 0.f};
    v8f acc1 = acc0;
    for (int k0 = 0; k0 < 64; k0 += 4) {
      const int k = k0 + 2 * kh;
      v2f a, b0, b1;
      a[0]  = V[(ti * 16 + l16) * LDG + k    ] * gg[k];     // (V diag g)[i][k]
      a[1]  = V[(ti * 16 + l16) * LDG + k + 1] * gg[k + 1];
      b0[0] = V[(tj0 * 16 + l16) * LDG + k    ];            // V^T[k][j] = V[j][k]
      b0[1] = V[(tj0 * 16 + l16) * LDG + k + 1];
      b1[0] = V[((tj0 + 1) * 16 + l16) * LDG + k    ];
      b1[1] = V[((tj0 + 1) * 16 + l16) * LDG + k + 1];
      acc0 = wmma4(a, b0, acc0);
      acc1 = wmma4(a, b1, acc1);
    }
#pragma unroll
    for (int v = 0; v < 8; ++v) {
      const int row = ti * 16 + v + 8 * kh;
      G[row * LDG + tj0 * 16 + l16]       = acc0[v];
      G[row * LDG + (tj0 + 1) * 16 + l16] = acc1[v];
    }
  }
  __syncthreads();

  // ---- 7) Y = Xc * M + m, * weight, store. Wave owns row-tiles 2w,2w+1;
  //          A fragment loaded once per K-step, reused across 4 col tiles. --
  const float w = wgt_sh;
#pragma unroll
  for (int th = 0; th < 2; ++th) {
    const int ti = wave * 2 + th;  // 0..15
    v8f acc[4];
#pragma unroll
    for (int j = 0; j < 4; ++j) acc[j] = (v8f){0.f,0.f,0.f,0.f,0.f,0.f,0.f,0.f};
    for (int k0 = 0; k0 < 64; k0 += 4) {
      const int k = k0 + 2 * kh;
      v2f a;
      a[0] = Xc[(ti * 16 + l16) * LDX + k    ];
      a[1] = Xc[(ti * 16 + l16) * LDX + k + 1];
#pragma unroll
      for (int tj = 0; tj < 4; ++tj) {
        v2f bf;
        bf[0] = G[(k    ) * LDG + tj * 16 + l16];
        bf[1] = G[(k + 1) * LDG + tj * 16 + l16];
        acc[tj] = wmma4(a, bf, acc[tj]);
      }
    }
#pragma unroll
    for (int tj = 0; tj < 4; ++tj) {
#pragma unroll
      for (int v = 0; v < 8; ++v) {
        const int row = ti * 16 + v + 8 * kh;
        const int col = tj * 16 + l16;
        out_x[(size_t)b * NROW * TCOL + row * TCOL + col] =
            (acc[tj][v] + mrow[row]) * w;
      }
    }
  }
  if (tid == 0) out_w[b] = w;
}

extern "C" void kernel_launch(void* const* d_in, const int* in_sizes, int n_in,
                              void* d_out, int out_size, void* d_ws,
                              size_t ws_size, hipStream_t stream) {
  (void)in_sizes; (void)n_in; (void)d_ws; (void)ws_size; (void)out_size;
  const float* x = (const float*)d_in[0];
  // d_in[1] (idx) is unused by the reference computation.
  float* out_x = (float*)d_out;
  float* out_w = out_x + (size_t)B_BATCH * NROW * TCOL;
  const size_t shmem =
      (size_t)(NROW * LDX + 2 * TCOL * LDG + NROW) * sizeof(float);
  hipLaunchKernelGGL(svd_denoise_kernel, dim3(B_BATCH), dim3(256), shmem,
                     stream, x, out_x, out_w);
}